// GeometricStructureEmbedding_2791728742883
// MI455X (gfx1250) — compile-verified
//
#include <hip/hip_runtime.h>
#include <hip/hip_bf16.h>
#include <math.h>

#define B_ 2
#define N_ 1024
#define SIGMA_D_INV 5.0f
#define FACTOR_A_ 3.8197186342054885f   /* 180/(15*pi) */
#define EPS_ 1e-6f

typedef __attribute__((ext_vector_type(2))) float v2f;
typedef __attribute__((ext_vector_type(8))) float v8f;

__device__ __forceinline__ float fsqrt_fast(float x) { return __builtin_amdgcn_sqrtf(x); }
__device__ __forceinline__ float frcp_fast(float x)  { return __builtin_amdgcn_rcpf(x); }

// atan2(s, c) specialized for s >= 0 (result in [0, pi]); atan2(0,0) -> 0.
// Cephes-style: ratio a = min/max in [0,1]; if a > tan(pi/8) reduce via
// atan(a) = pi/4 + atan((a-1)/(a+1)); 4-term odd minimax poly on the rest.
__device__ __forceinline__ float atan2_pos_fast(float s, float c) {
  const float ac = fabsf(c);
  const float mx = fmaxf(s, ac);
  const float mn = fminf(s, ac);
  float a = mn * frcp_fast(mx);                    // [0,1] (nan if s==c==0)
  const bool red = a > 0.41421356f;                // tan(pi/8)
  float x = red ? (a - 1.0f) * frcp_fast(a + 1.0f) : a;
  const float z = x * x;
  float p = fmaf(8.05374449538e-2f, z, -1.38776856032e-1f);
  p = fmaf(p, z, 1.99777106478e-1f);
  p = fmaf(p, z, -3.33329491539e-1f);
  float r = fmaf(p * z, x, x);                     // atan(x)
  if (red) r += 0.78539816340f;                    // + pi/4
  if (s > ac) r = 1.57079632679f - r;              // swap -> pi/2 - atan
  if (c < 0.0f) r = 3.14159265359f - r;            // quadrant II
  if (mx == 0.0f) r = 0.0f;                        // atan2(0,0) = 0
  return r;
}

// ---------------------------------------------------------------------------
// Kernel 1: per-row top-8 nearest neighbors (excluding self) -> ref_vec in ws
// grid: B * (N/16) blocks, 256 threads (16 lanes per row, 16 rows per block)
// ---------------------------------------------------------------------------
__global__ __launch_bounds__(256) void gse_knn_kernel(
    const float* __restrict__ points, float* __restrict__ refv /* [B][N][8][3] */) {
  __shared__ float px[N_], py[N_], pz[N_], sq[N_];
  __shared__ float cd[256 * 8];
  __shared__ int   ci[256 * 8];

  const int tid = threadIdx.x;
  const int b   = blockIdx.x / (N_ / 16);
  const int ib  = blockIdx.x % (N_ / 16);

  for (int t = tid; t < N_; t += 256) {
    const size_t base = ((size_t)b * N_ + t) * 3;
    float x = points[base + 0], y = points[base + 1], z = points[base + 2];
    px[t] = x; py[t] = y; pz[t] = z;
    sq[t] = x * x + y * y + z * z;
  }
  __syncthreads();

  const int lane16 = tid & 15;
  const int il     = tid >> 4;            // local row 0..15
  const int ig     = ib * 16 + il;        // global row
  const float xi = px[ig], yi = py[ig], zi = pz[ig], sqi = sq[ig];

  float bd[8]; int bix[8];
#pragma unroll
  for (int k = 0; k < 8; ++k) { bd[k] = 3.4e38f; bix[k] = 0x7FFFFFFF; }

  for (int j = lane16; j < N_; j += 16) {
    if (j == ig) continue;
    float d = sqi + sq[j] - 2.0f * (xi * px[j] + yi * py[j] + zi * pz[j]);
    if (d < bd[7] || (d == bd[7] && j < bix[7])) {
      bd[7] = d; bix[7] = j;
#pragma unroll
      for (int k = 6; k >= 0; --k) {
        bool sw = (bd[k + 1] < bd[k]) || (bd[k + 1] == bd[k] && bix[k + 1] < bix[k]);
        if (sw) {
          float td = bd[k]; bd[k] = bd[k + 1]; bd[k + 1] = td;
          int ti = bix[k]; bix[k] = bix[k + 1]; bix[k + 1] = ti;
        }
      }
    }
  }
#pragma unroll
  for (int k = 0; k < 8; ++k) { cd[tid * 8 + k] = bd[k]; ci[tid * 8 + k] = bix[k]; }
  __syncthreads();

  if (tid < 16) {
    float md[8]; int mi[8];
#pragma unroll
    for (int k = 0; k < 8; ++k) { md[k] = 3.4e38f; mi[k] = 0x7FFFFFFF; }
    const int base = tid * 128;           // 16 threads * 8 entries each
    for (int c = 0; c < 128; ++c) {
      float d = cd[base + c]; int j = ci[base + c];
      if (d < md[7] || (d == md[7] && j < mi[7])) {
        md[7] = d; mi[7] = j;
#pragma unroll
        for (int k = 6; k >= 0; --k) {
          bool sw = (md[k + 1] < md[k]) || (md[k + 1] == md[k] && mi[k + 1] < mi[k]);
          if (sw) {
            float td = md[k]; md[k] = md[k + 1]; md[k + 1] = td;
            int ti = mi[k]; mi[k] = mi[k + 1]; mi[k + 1] = ti;
          }
        }
      }
    }
    const int iglob = ib * 16 + tid;
#pragma unroll
    for (int k = 0; k < 8; ++k) {
      const int nb = mi[k];
      const size_t rb = (((size_t)b * N_ + iglob) * 8 + k) * 3;
      refv[rb + 0] = px[nb] - px[iglob];
      refv[rb + 1] = py[nb] - py[iglob];
      refv[rb + 2] = pz[nb] - pz[iglob];
    }
  }
}

// ---------------------------------------------------------------------------
// Kernel 2: tile pass. Each wave computes one 16x16 tile via 4x
// v_wmma_f32_16x16x4_f32, then the per-pair transcendentals + (i,j,k) atan2s.
// grid: B * 64 * 8 blocks (jc = 128-col chunk), 256 threads = 8 waves.
// ---------------------------------------------------------------------------
__global__ __launch_bounds__(256) void gse_tile_kernel(
    const float* __restrict__ points, const float* __restrict__ normals,
    const float* __restrict__ refv, float* __restrict__ out) {
  __shared__ float jx[128], jy[128], jz[128];
  __shared__ float jnx[128], jny[128], jnz[128];
  __shared__ float jsq[128], jnp[128], jnn[128];
  __shared__ float ixs[16], iys[16], izs[16];
  __shared__ float inx[16], iny[16], inz[16];
  __shared__ float isq[16], inp[16], inn[16];
  __shared__ float rvx[128], rvy[128], rvz[128];   // [16 rows][8 nbrs]

  const int tid = threadIdx.x;
  const int jc = blockIdx.x & 7;
  const int ib = (blockIdx.x >> 3) & 63;
  const int b  = blockIdx.x >> 9;

  if (tid < 128) {
    const int jg = jc * 128 + tid;
    const size_t base = ((size_t)b * N_ + jg) * 3;
    float x = points[base + 0], y = points[base + 1], z = points[base + 2];
    float nx = normals[base + 0], ny = normals[base + 1], nz = normals[base + 2];
    jx[tid] = x; jy[tid] = y; jz[tid] = z;
    jnx[tid] = nx; jny[tid] = ny; jnz[tid] = nz;
    jsq[tid] = x * x + y * y + z * z;
    jnp[tid] = nx * x + ny * y + nz * z;
    jnn[tid] = fsqrt_fast(nx * nx + ny * ny + nz * nz);
  } else if (tid < 144) {
    const int il = tid - 128;
    const int iglob = ib * 16 + il;
    const size_t base = ((size_t)b * N_ + iglob) * 3;
    float x = points[base + 0], y = points[base + 1], z = points[base + 2];
    float nx = normals[base + 0], ny = normals[base + 1], nz = normals[base + 2];
    ixs[il] = x; iys[il] = y; izs[il] = z;
    inx[il] = nx; iny[il] = ny; inz[il] = nz;
    isq[il] = x * x + y * y + z * z;
    inp[il] = nx * x + ny * y + nz * z;
    inn[il] = fsqrt_fast(nx * nx + ny * ny + nz * nz);
  }
  if (tid < 128) {   // 16 rows x 8 neighbors of ref_vec
    const int il = tid >> 3, k = tid & 7;
    const size_t rb = (((size_t)b * N_ + ib * 16 + il) * 8 + k) * 3;
    rvx[tid] = refv[rb + 0]; rvy[tid] = refv[rb + 1]; rvz[tid] = refv[rb + 2];
  }
  __syncthreads();

  const int lane = tid & 31;
  const int w    = tid >> 5;              // wave -> j-tile within chunk
  const int n    = lane & 15;             // tile column for B/C/D
  const int hi   = lane >> 4;             // K-half select for A/B
  const int m0   = lane & 15;             // A row
  const int jl   = w * 16 + n;            // LDS col index 0..127
  const int jg   = jc * 128 + jl;         // global column

  // A fragments: lane holds row m0, K = 2*hi + {component 0, component 1}
  v2f a_pts = { hi ? izs[m0] : ixs[m0], hi ? isq[m0]   : iys[m0] };  // [x,y,z,sq_i]
  v2f a_dot = { hi ? inz[m0] : inx[m0], hi ? -inp[m0]  : iny[m0] };  // [n_i, -n_i.p_i]
  v2f a_tr  = { hi ? izs[m0] : ixs[m0], hi ? 1.0f      : iys[m0] };  // [p_i, 1]
  v2f a_nn  = { hi ? inz[m0] : inx[m0], hi ? 0.0f      : iny[m0] };  // [n_i, 0]
  // B fragments: lane holds col jl, K = 2*hi + {component 0, component 1}
  v2f b_d2  = { hi ? -2.0f * jz[jl] : -2.0f * jx[jl], hi ? 1.0f     : -2.0f * jy[jl] };
  v2f b_dot = { hi ? jz[jl]  : jx[jl],  hi ? 1.0f     : jy[jl] };    // [p_j, 1]
  v2f b_tr  = { hi ? jnz[jl] : jnx[jl], hi ? -jnp[jl] : jny[jl] };   // [n_j, -n_j.p_j]
  v2f b_nn  = { hi ? jnz[jl] : jnx[jl], hi ? 0.0f     : jny[jl] };   // [n_j, 0]

  const float sqj = jsq[jl];
  v8f c_d2 = { sqj, sqj, sqj, sqj, sqj, sqj, sqj, sqj };
  v8f zero = { 0.f, 0.f, 0.f, 0.f, 0.f, 0.f, 0.f, 0.f };

  // d2 = sq_i + sq_j - 2 p_i.p_j ; dij = n_i.(p_j-p_i) ; dji = n_j.(p_i-p_j) ; nn = n_i.n_j
  v8f acc_d2 = __builtin_amdgcn_wmma_f32_16x16x4_f32(false, a_pts, false, b_d2, (short)0, c_d2, false, false);
  v8f acc_ij = __builtin_amdgcn_wmma_f32_16x16x4_f32(false, a_dot, false, b_dot, (short)0, zero, false, false);
  v8f acc_ji = __builtin_amdgcn_wmma_f32_16x16x4_f32(false, a_tr,  false, b_tr,  (short)0, zero, false, false);
  v8f acc_nn = __builtin_amdgcn_wmma_f32_16x16x4_f32(false, a_nn,  false, b_nn,  (short)0, zero, false, false);

  const float pjx = jx[jl], pjy = jy[jl], pjz = jz[jl];
  const float nnj = jnn[jl];
  const size_t aoff = (size_t)B_ * N_ * N_ * 3;   // start of a_indices in d_out

#pragma unroll
  for (int v = 0; v < 8; ++v) {
    const int mrow = v + 8 * hi;                  // D layout: row = vgpr + 8*(lane/16)
    const int iglob = ib * 16 + mrow;

    const float ax = pjx - ixs[mrow];
    const float ay = pjy - iys[mrow];
    const float az = pjz - izs[mrow];
    const float lnorm = fsqrt_fast(ax * ax + ay * ay + az * az);

    const float dist = fsqrt_fast(fmaxf(acc_d2[v], 0.0f));
    float d_idx = dist * SIGMA_D_INV;

    float c1 = acc_ij[v] * frcp_fast(inn[mrow] * lnorm + EPS_);
    c1 = fminf(fmaxf(c1, -1.0f), 1.0f);
    float c2 = acc_ji[v] * frcp_fast(nnj * lnorm + EPS_);
    c2 = fminf(fmaxf(c2, -1.0f), 1.0f);
    float ang = fabsf(acosf(c1) - acosf(c2));

    float xq = acc_nn[v] * frcp_fast(inn[mrow] * nnj);
    if (xq != xq) xq = 0.0f;
    xq = fminf(fmaxf(xq, -1.0f), 1.0f);
    const float seta = acosf(xq);

    if (iglob == jg) { d_idx = 0.0f; ang = 0.0f; }

    const size_t base3 = (((size_t)b * N_ + iglob) * N_ + jg) * 3;
    out[base3 + 0] = d_idx;
    out[base3 + 1] = seta;
    out[base3 + 2] = ang;

    float vals[8];
#pragma unroll
    for (int k = 0; k < 8; ++k) {
      const float rx = rvx[mrow * 8 + k], ry = rvy[mrow * 8 + k], rz = rvz[mrow * 8 + k];
      const float cx = ry * az - rz * ay;
      const float cy = rz * ax - rx * az;
      const float cz = rx * ay - ry * ax;
      const float s = fsqrt_fast(cx * cx + cy * cy + cz * cz);
      const float c = rx * ax + ry * ay + rz * az;
      vals[k] = atan2_pos_fast(s, c) * FACTOR_A_;
    }
    const size_t base8 = aoff + (((size_t)b * N_ + iglob) * N_ + jg) * 8;
    float4* o4 = reinterpret_cast<float4*>(out + base8);
    o4[0] = make_float4(vals[0], vals[1], vals[2], vals[3]);
    o4[1] = make_float4(vals[4], vals[5], vals[6], vals[7]);
  }
}

extern "C" void kernel_launch(void* const* d_in, const int* in_sizes, int n_in,
                              void* d_out, int out_size, void* d_ws, size_t ws_size,
                              hipStream_t stream) {
  (void)in_sizes; (void)n_in; (void)out_size; (void)ws_size;
  const float* points  = (const float*)d_in[0];
  const float* normals = (const float*)d_in[1];
  float* out  = (float*)d_out;
  float* refv = (float*)d_ws;   // B*N*8*3 floats = 192 KB of scratch

  gse_knn_kernel<<<dim3(B_ * (N_ / 16)), dim3(256), 0, stream>>>(points, refv);
  gse_tile_kernel<<<dim3(B_ * 64 * 8), dim3(256), 0, stream>>>(points, normals, refv, out);
}